// NeuroSATLayer_27144193311187
// MI455X (gfx1250) — compile-verified
//
#include <hip/hip_runtime.h>
#include <math.h>

// ---------------------------------------------------------------------------
// NeuroSAT message-passing round for MI455X (gfx1250, wave32, WMMA).
// All GEMMs run on V_WMMA_F32_16X16X4_F32 (fp32 in / fp32 accum) to match the
// fp32 reference numerics. Weights (~1.6MB) stay L2-resident; activations are
// tiled through LDS (128-row tiles, padded stride).
// ---------------------------------------------------------------------------

#define EMB        128
#define N_VARS     65536
#define N_LITS     131072
#define N_CLAUSES  131072
#define N_PAIRS    (N_CLAUSES * 4)
#define ROWS       128            // rows per workgroup tile (8 waves x 16)
#define LDA        (EMB + 4)      // padded LDS row stride (floats)
#define LDA2       (2 * EMB + 4)  // padded stride for 256-wide LSTM input

typedef __attribute__((ext_vector_type(2))) float v2f;
typedef __attribute__((ext_vector_type(8))) float v8f;

// Accumulate a 16x16 fp32 tile of  A(16 x K) * W^T  (W is N x K row-major,
// pre-offset to its 16-row slice).  Per ISA 7.12.2:
//   A frag: lane m = lane&15, VGPR0 = K=kb+2*hi, VGPR1 = K=kb+1+2*hi
//   B frag: lane n = lane&15, same K striping (B row striped across lanes)
//   C/D   : VGPR r -> M = r + 8*hi, N = lane&15
__device__ __forceinline__ v8f wmma_xwt(const float* __restrict__ A, int lda,
                                        const float* __restrict__ W, int ldw,
                                        int K, int m, int hi, v8f acc) {
#pragma unroll 8
  for (int kb = 0; kb < K; kb += 4) {
    v2f a, b;
    a.x = A[m * lda + kb + 2 * hi];
    a.y = A[m * lda + kb + 2 * hi + 1];
    b.x = W[m * ldw + kb + 2 * hi];
    b.y = W[m * ldw + kb + 2 * hi + 1];
    acc = __builtin_amdgcn_wmma_f32_16x16x4_f32(false, a, false, b,
                                                (short)0, acc, false, false);
  }
  return acc;
}

__device__ __forceinline__ float sigmoidf_(float x) {
  return 1.0f / (1.0f + __expf(-x));
}

// ---------------------------------------------------------------------------
// Kernel 1/3: fused 3-layer MLP  Y = L2(relu(L1(relu(L0(X)))))
//   X: [nrows,128]  W: [3,128,128]  B: [3,128]
// Layer loop force-unrolled so each epilogue is straight-line (no runtime
// relu / destination branching).
// ---------------------------------------------------------------------------
__global__ __launch_bounds__(256) void mlp3_kernel(
    const float* __restrict__ X, const float* __restrict__ W,
    const float* __restrict__ Bv, float* __restrict__ Y) {
  __shared__ float buf0[ROWS * LDA];
  __shared__ float buf1[ROWS * LDA];
  const int row0 = blockIdx.x * ROWS;
  const int tid  = threadIdx.x;

  // cooperative load of the 128x128 activation tile
  for (int i = tid; i < ROWS * (EMB / 4); i += 256) {
    const int r  = i >> 5;           // EMB/4 == 32
    const int c4 = (i & 31) << 2;
    *(float4*)(buf0 + r * LDA + c4) =
        *(const float4*)(X + (size_t)(row0 + r) * EMB + c4);
  }
  __syncthreads();

  const int wave = tid >> 5, lane = tid & 31;
  const int m = lane & 15, hi = lane >> 4;

#pragma unroll
  for (int layer = 0; layer < 3; ++layer) {
    float* cur = (layer & 1) ? buf1 : buf0;
    float* nxt = (layer & 1) ? buf0 : buf1;
    const float* Wl = W + (size_t)layer * EMB * EMB;
    const float* bl = Bv + layer * EMB;
    const float* Arow = cur + (wave * 16) * LDA;
    for (int ct = 0; ct < 8; ++ct) {
      v8f acc = {};
      acc = wmma_xwt(Arow, LDA, Wl + (size_t)(ct * 16) * EMB, EMB, EMB, m, hi, acc);
      const int col  = ct * 16 + m;
      const float bb = bl[col];
      if (layer < 2) {
#pragma unroll
        for (int r = 0; r < 8; ++r) {
          const int rl = wave * 16 + r + 8 * hi;
          nxt[rl * LDA + col] = fmaxf(acc[r] + bb, 0.0f);
        }
      } else {
#pragma unroll
        for (int r = 0; r < 8; ++r) {
          const int rl = wave * 16 + r + 8 * hi;
          Y[(size_t)(row0 + rl) * EMB + col] = acc[r] + bb;
        }
      }
    }
    __syncthreads();
  }
}

// ---------------------------------------------------------------------------
// Kernel 2: clause side.  lc_msg[c] = sum of 4 gathered l_pre rows, then
// LSTM gates = lc_msg*Wih^T + c_h*Whh^T + b, fused i/f/g/o epilogue.
// ---------------------------------------------------------------------------
__global__ __launch_bounds__(256) void clause_lstm_kernel(
    const float* __restrict__ l_pre, const int* __restrict__ lit,
    const float* __restrict__ ch, const float* __restrict__ cc,
    const float* __restrict__ Wih, const float* __restrict__ Whh,
    const float* __restrict__ bih, const float* __restrict__ bhh,
    float* __restrict__ h2out, float* __restrict__ c2out) {
  __shared__ float xbuf[ROWS * LDA];
  const int row0 = blockIdx.x * ROWS;
  const int tid  = threadIdx.x;

  // gather-sum: lc_msg row r = sum_j l_pre[lit[4*(row0+r)+j]]
  for (int i = tid; i < ROWS * (EMB / 4); i += 256) {
    const int r  = i >> 5;
    const int c4 = (i & 31) << 2;
    const int p  = (row0 + r) << 2;
    const float4 v0 = *(const float4*)(l_pre + (size_t)lit[p + 0] * EMB + c4);
    const float4 v1 = *(const float4*)(l_pre + (size_t)lit[p + 1] * EMB + c4);
    const float4 v2 = *(const float4*)(l_pre + (size_t)lit[p + 2] * EMB + c4);
    const float4 v3 = *(const float4*)(l_pre + (size_t)lit[p + 3] * EMB + c4);
    float4 s;
    s.x = v0.x + v1.x + v2.x + v3.x;
    s.y = v0.y + v1.y + v2.y + v3.y;
    s.z = v0.z + v1.z + v2.z + v3.z;
    s.w = v0.w + v1.w + v2.w + v3.w;
    *(float4*)(xbuf + r * LDA + c4) = s;
  }
  __syncthreads();

  const int wave = tid >> 5, lane = tid & 31;
  const int m = lane & 15, hi = lane >> 4;
  const float* Arow = xbuf + (wave * 16) * LDA;
  const float* Hrow = ch + (size_t)(row0 + wave * 16) * EMB;

  for (int ct = 0; ct < 8; ++ct) {
    v8f g[4];
#pragma unroll
    for (int gate = 0; gate < 4; ++gate) {
      const float* Wt = Wih + (size_t)(gate * EMB + ct * 16) * EMB;
      const float* Ut = Whh + (size_t)(gate * EMB + ct * 16) * EMB;
      v8f acc = {};
      acc = wmma_xwt(Arow, LDA, Wt, EMB, EMB, m, hi, acc);   // x @ Wih^T
      acc = wmma_xwt(Hrow, EMB, Ut, EMB, EMB, m, hi, acc);   // h @ Whh^T
      g[gate] = acc;
    }
    const int col = ct * 16 + m;
    const float bi = bih[0 * EMB + col] + bhh[0 * EMB + col];
    const float bf = bih[1 * EMB + col] + bhh[1 * EMB + col];
    const float bg = bih[2 * EMB + col] + bhh[2 * EMB + col];
    const float bo = bih[3 * EMB + col] + bhh[3 * EMB + col];
#pragma unroll
    for (int r = 0; r < 8; ++r) {
      const size_t row = (size_t)(row0 + wave * 16 + r + 8 * hi);
      const float cprev = cc[row * EMB + col];
      const float iv = sigmoidf_(g[0][r] + bi);
      const float fv = sigmoidf_(g[1][r] + bf);
      const float gv = tanhf(g[2][r] + bg);
      const float ov = sigmoidf_(g[3][r] + bo);
      const float cn = fv * cprev + iv * gv;
      c2out[row * EMB + col] = cn;
      h2out[row * EMB + col] = ov * tanhf(cn);
    }
  }
}

// ---------------------------------------------------------------------------
// Kernel 4a/4b: zero cl_msg, then scatter-add c_pre rows into literal rows.
// ---------------------------------------------------------------------------
__global__ void zero_kernel(float* __restrict__ p) {
  const size_t i = (size_t)blockIdx.x * blockDim.x + threadIdx.x;
  float4 z; z.x = z.y = z.z = z.w = 0.0f;
  *(float4*)(p + i * 4) = z;
}

__global__ void scatter_kernel(const float* __restrict__ c_pre,
                               const int* __restrict__ lit,
                               float* __restrict__ cl_msg) {
  const size_t idx = (size_t)blockIdx.x * blockDim.x + threadIdx.x;
  const int p  = (int)(idx >> 5);          // pair index
  const int c4 = (int)(idx & 31) << 2;     // float4 column
  const int dst = lit[p];
  const float4 v = *(const float4*)(c_pre + (size_t)(p >> 2) * EMB + c4);
  float* o = cl_msg + (size_t)dst * EMB + c4;
  atomicAdd(o + 0, v.x);
  atomicAdd(o + 1, v.y);
  atomicAdd(o + 2, v.z);
  atomicAdd(o + 3, v.w);
}

// ---------------------------------------------------------------------------
// Kernel 5: literal LSTM.  x = [cl_msg || l_pre_flipped]  (K=256 for Wih).
// ---------------------------------------------------------------------------
__global__ __launch_bounds__(256) void lit_lstm_kernel(
    const float* __restrict__ cl_msg, const float* __restrict__ l_pre,
    const float* __restrict__ lh, const float* __restrict__ lc,
    const float* __restrict__ Wih, const float* __restrict__ Whh,
    const float* __restrict__ bih, const float* __restrict__ bhh,
    float* __restrict__ h2out, float* __restrict__ c2out) {
  __shared__ float xbuf[ROWS * LDA2];
  const int row0 = blockIdx.x * ROWS;
  const int tid  = threadIdx.x;

  // build l_in rows: cols [0,128) = cl_msg, cols [128,256) = l_pre[flip(row)]
  for (int i = tid; i < ROWS * (2 * EMB / 4); i += 256) {
    const int r  = i >> 6;                // 2*EMB/4 == 64
    const int c4 = (i & 63) << 2;
    const int row = row0 + r;
    float4 v;
    if (c4 < EMB) {
      v = *(const float4*)(cl_msg + (size_t)row * EMB + c4);
    } else {
      const int fr = (row + N_VARS) & (N_LITS - 1);
      v = *(const float4*)(l_pre + (size_t)fr * EMB + (c4 - EMB));
    }
    *(float4*)(xbuf + r * LDA2 + c4) = v;
  }
  __syncthreads();

  const int wave = tid >> 5, lane = tid & 31;
  const int m = lane & 15, hi = lane >> 4;
  const float* Arow = xbuf + (wave * 16) * LDA2;
  const float* Hrow = lh + (size_t)(row0 + wave * 16) * EMB;

  for (int ct = 0; ct < 8; ++ct) {
    v8f g[4];
#pragma unroll
    for (int gate = 0; gate < 4; ++gate) {
      const float* Wt = Wih + (size_t)(gate * EMB + ct * 16) * (2 * EMB);
      const float* Ut = Whh + (size_t)(gate * EMB + ct * 16) * EMB;
      v8f acc = {};
      acc = wmma_xwt(Arow, LDA2, Wt, 2 * EMB, 2 * EMB, m, hi, acc);  // K=256
      acc = wmma_xwt(Hrow, EMB, Ut, EMB, EMB, m, hi, acc);           // K=128
      g[gate] = acc;
    }
    const int col = ct * 16 + m;
    const float bi = bih[0 * EMB + col] + bhh[0 * EMB + col];
    const float bf = bih[1 * EMB + col] + bhh[1 * EMB + col];
    const float bg = bih[2 * EMB + col] + bhh[2 * EMB + col];
    const float bo = bih[3 * EMB + col] + bhh[3 * EMB + col];
#pragma unroll
    for (int r = 0; r < 8; ++r) {
      const size_t row = (size_t)(row0 + wave * 16 + r + 8 * hi);
      const float cprev = lc[row * EMB + col];
      const float iv = sigmoidf_(g[0][r] + bi);
      const float fv = sigmoidf_(g[1][r] + bf);
      const float gv = tanhf(g[2][r] + bg);
      const float ov = sigmoidf_(g[3][r] + bo);
      const float cn = fv * cprev + iv * gv;
      c2out[row * EMB + col] = cn;
      h2out[row * EMB + col] = ov * tanhf(cn);
    }
  }
}

// ---------------------------------------------------------------------------
extern "C" void kernel_launch(void* const* d_in, const int* in_sizes, int n_in,
                              void* d_out, int out_size, void* d_ws, size_t ws_size,
                              hipStream_t stream) {
  (void)in_sizes; (void)n_in; (void)out_size; (void)ws_size;
  const float* l_h    = (const float*)d_in[0];
  const float* l_c    = (const float*)d_in[1];
  const float* c_h    = (const float*)d_in[2];
  const float* c_c    = (const float*)d_in[3];
  const float* Lmsg_W = (const float*)d_in[4];
  const float* Lmsg_b = (const float*)d_in[5];
  const float* Cmsg_W = (const float*)d_in[6];
  const float* Cmsg_b = (const float*)d_in[7];
  const float* Lu_Wih = (const float*)d_in[8];
  const float* Lu_Whh = (const float*)d_in[9];
  const float* Lu_bih = (const float*)d_in[10];
  const float* Lu_bhh = (const float*)d_in[11];
  const float* Cu_Wih = (const float*)d_in[12];
  const float* Cu_Whh = (const float*)d_in[13];
  const float* Cu_bih = (const float*)d_in[14];
  const float* Cu_bhh = (const float*)d_in[15];
  const int*   lit    = (const int*)d_in[17];   // edge_src[0:N_PAIRS]

  float* out  = (float*)d_out;
  float* l_h2 = out + 0 * (size_t)N_LITS * EMB;
  float* l_c2 = out + 1 * (size_t)N_LITS * EMB;
  float* c_h2 = out + 2 * (size_t)N_LITS * EMB;
  float* c_c2 = out + 3 * (size_t)N_LITS * EMB;

  float* l_pre  = (float*)d_ws;                               // 64 MB
  float* c_pre  = l_pre + (size_t)N_LITS * EMB;               // 64 MB
  float* cl_msg = c_pre + (size_t)N_CLAUSES * EMB;            // 64 MB

  // 1) literal message MLP
  mlp3_kernel<<<N_LITS / ROWS, 256, 0, stream>>>(l_h, Lmsg_W, Lmsg_b, l_pre);
  // 2) clause gather + clause LSTM (writes c_h2/c_c2 directly to output)
  clause_lstm_kernel<<<N_CLAUSES / ROWS, 256, 0, stream>>>(
      l_pre, lit, c_h, c_c, Cu_Wih, Cu_Whh, Cu_bih, Cu_bhh, c_h2, c_c2);
  // 3) clause message MLP
  mlp3_kernel<<<N_CLAUSES / ROWS, 256, 0, stream>>>(c_h2, Cmsg_W, Cmsg_b, c_pre);
  // 4) scatter clause messages to literals
  zero_kernel<<<(N_LITS * EMB / 4) / 256, 256, 0, stream>>>(cl_msg);
  scatter_kernel<<<((size_t)N_PAIRS * 32) / 256, 256, 0, stream>>>(c_pre, lit, cl_msg);
  // 5) literal LSTM (writes l_h2/l_c2 to output)
  lit_lstm_kernel<<<N_LITS / ROWS, 256, 0, stream>>>(
      cl_msg, l_pre, l_h, l_c, Lu_Wih, Lu_Whh, Lu_bih, Lu_bhh, l_h2, l_c2);
}